// KNN_50345606644134
// MI455X (gfx1250) — compile-verified
//
#include <hip/hip_runtime.h>

typedef __attribute__((ext_vector_type(2))) float v2f;
typedef __attribute__((ext_vector_type(8))) float v8f;

#define NB      2
#define NPTS    8192
#define DIM     128
#define K_OUT   17
#define ROWS    128      // query rows per block
#define CHUNK   16       // candidate columns per iteration
#define BSTRIDE 132      // padded K stride for B tile in LDS (avoids bank conflicts)
#define DSTRIDE 17       // padded col stride for dist tile in LDS

// ---------------- squared-norm precompute ----------------
__global__ __launch_bounds__(256) void sqnorm_kernel(const float* __restrict__ x,
                                                     float* __restrict__ sq) {
  int row = blockIdx.x * blockDim.x + threadIdx.x;
  if (row < NB * NPTS) {
    const float4* xr = (const float4*)(x + (size_t)row * DIM);
    float s = 0.f;
#pragma unroll
    for (int i = 0; i < DIM / 4; ++i) {
      float4 v = xr[i];
      s += v.x * v.x + v.y * v.y + v.z * v.z + v.w * v.w;
    }
    sq[row] = s;
  }
}

// ---------------- fused WMMA distance + top-k ----------------
__global__ __launch_bounds__(256) void knn_kernel(const float* __restrict__ x,
                                                  const float* __restrict__ sq,
                                                  float* __restrict__ outD,
                                                  float* __restrict__ outI) {
  __shared__ float Bld[CHUNK * BSTRIDE];   // B tile: [col][k], padded
  __shared__ float Dld[ROWS * DSTRIDE];    // -2*dot tile: [row][col], padded
  __shared__ float SqB[CHUNK];

  const int tid   = threadIdx.x;
  const int wave  = tid >> 5;
  const int lane  = tid & 31;
  const int lhalf = lane >> 4;          // 0: lanes 0-15, 1: lanes 16-31
  const int l16   = lane & 15;
  const int koff  = lhalf * 2;          // K offset of this half per WMMA fragment layout

  const int blocksPerBatch = NPTS / ROWS;                 // 64
  const int b       = blockIdx.x / blocksPerBatch;
  const int rowBase = (blockIdx.x % blocksPerBatch) * ROWS;

  const float* xb  = x  + (size_t)b * NPTS * DIM;
  const float* sqb = sq + (size_t)b * NPTS;

  // A fragments in registers, pre-scaled by -2 so WMMA accumulates -2*dot.
  // 16x4 f32 A layout: lane holds row M=l16; VGPR0 = K=koff, VGPR1 = K=koff+1.
  const int arow = rowBase + wave * 16 + l16;
  float a[2 * (DIM / 4)];
  {
    const float* ap = xb + (size_t)arow * DIM + koff;
#pragma unroll
    for (int s = 0; s < DIM / 4; ++s) {
      v2f t = *(const v2f*)(ap + 4 * s);
      a[2 * s]     = -2.0f * t.x;
      a[2 * s + 1] = -2.0f * t.y;
    }
  }

  // Per-query top-k state (threads 0..ROWS-1 each own one query row)
  float bd[K_OUT];
  int   bi[K_OUT];
#pragma unroll
  for (int j = 0; j < K_OUT; ++j) { bd[j] = 3.0e38f; bi[j] = 0x7fffffff; }
  float sqA = 0.f;
  int   qi  = 0;
  if (tid < ROWS) { qi = rowBase + tid; sqA = sqb[qi]; }

  for (int c0 = 0; c0 < NPTS; c0 += CHUNK) {
    // ---- stage B tile (16 candidate rows x 128 dims) into LDS ----
    {
      int col = tid >> 4;            // 0..15
      int k0  = (tid & 15) * 8;      // 0..120
      const float4* src = (const float4*)(xb + (size_t)(c0 + col) * DIM + k0);
      float4 v0 = src[0];
      float4 v1 = src[1];
      float* dst = &Bld[col * BSTRIDE + k0];
      *(float4*)(dst)     = v0;
      *(float4*)(dst + 4) = v1;
      if (tid < CHUNK) SqB[tid] = sqb[c0 + tid];
    }
    __syncthreads();

    // ---- 16x16 tile of -2*x.x^T via chained f32 WMMA over K=128 ----
    v8f acc = {};
    const float* bp = &Bld[l16 * BSTRIDE + koff];   // B frag: lane holds col N=l16
#pragma unroll
    for (int s = 0; s < DIM / 4; ++s) {
      v2f bfrag = *(const v2f*)(bp + 4 * s);
      v2f afrag;
      afrag.x = a[2 * s];
      afrag.y = a[2 * s + 1];
      acc = __builtin_amdgcn_wmma_f32_16x16x4_f32(
          /*neg_a=*/false, afrag, /*neg_b=*/false, bfrag,
          /*c_mod=*/(short)0, acc, /*reuse_a=*/false, /*reuse_b=*/false);
    }

    // ---- scatter C fragment to LDS: M = v + 8*lhalf, N = l16 ----
#pragma unroll
    for (int v = 0; v < 8; ++v) {
      int r = wave * 16 + v + 8 * lhalf;
      Dld[r * DSTRIDE + l16] = acc[v];
    }
    __syncthreads();

    // ---- top-k insertion: one thread per query row ----
    if (tid < ROWS) {
#pragma unroll 1
      for (int j = 0; j < CHUNK; ++j) {
        int   ci = c0 + j;
        float d2 = Dld[tid * DSTRIDE + j] + sqA + SqB[j];
        if (ci != qi && d2 < bd[K_OUT - 1]) {
          float dd = d2;
          int   ii = ci;
#pragma unroll
          for (int p = 0; p < K_OUT; ++p) {
            bool sw = dd < bd[p];   // strict < keeps stable (index-ascending) tie order
            float td = bd[p]; int ti = bi[p];
            if (sw) { bd[p] = dd; bi[p] = ii; dd = td; ii = ti; }
          }
        }
      }
    }
    __syncthreads();   // protects Bld/Dld before next iteration
  }

  if (tid < ROWS) {
    size_t o = ((size_t)b * NPTS + (size_t)qi) * K_OUT;
#pragma unroll
    for (int j = 0; j < K_OUT; ++j) {
      outD[o + j] = sqrtf(fmaxf(bd[j], 0.0f));
      outI[o + j] = (float)bi[j];
    }
  }
}

extern "C" void kernel_launch(void* const* d_in, const int* in_sizes, int n_in,
                              void* d_out, int out_size, void* d_ws, size_t ws_size,
                              hipStream_t stream) {
  const float* x  = (const float*)d_in[0];
  float* sqws     = (float*)d_ws;                 // 16384 floats
  float* out      = (float*)d_out;

  const size_t nOutVals = (size_t)NB * NPTS * K_OUT;   // 278528
  const size_t xVals    = (size_t)NB * NPTS * DIM;     // 2097152
  float* outD  = out;
  float* outI  = out + nOutVals;
  float* outX0 = out + 2 * nOutVals;
  float* outX1 = out + 2 * nOutVals + xVals;

  // 1) squared norms of all rows
  sqnorm_kernel<<<(NB * NPTS + 255) / 256, 256, 0, stream>>>(x, sqws);

  // 2) fused WMMA distances + top-k
  knn_kernel<<<NB * (NPTS / ROWS), 256, 0, stream>>>(x, sqws, outD, outI);

  // 3) pass-through copies of x (reference returns (dist, idx, x, x))
  hipMemcpyAsync(outX0, x, xVals * sizeof(float), hipMemcpyDeviceToDevice, stream);
  hipMemcpyAsync(outX1, x, xVals * sizeof(float), hipMemcpyDeviceToDevice, stream);
}